// SimpleGaussianRenderer_26560077758964
// MI455X (gfx1250) — compile-verified
//
#include <hip/hip_runtime.h>
#include <cmath>

#ifndef __has_builtin
#define __has_builtin(x) 0
#endif

#define GW 320
#define GH 240
#define GN 512
#define NPAR 10   // sx, sy, xi, yi, rad, coef(-0.5/ss^2), opEff, cr, cg, cb

typedef __attribute__((ext_vector_type(2))) float v2f;
typedef __attribute__((ext_vector_type(8))) float v8f;

// Exact pointee type from the builtin's declared signature:
//   "__attribute__((__vector_size__(4 * sizeof(int)))) int __device__ *"
typedef int v4i_raw __attribute__((vector_size(4 * sizeof(int))));
typedef __attribute__((address_space(1))) v4i_raw gv4i;   // global (__device__)
typedef __attribute__((address_space(3))) v4i_raw lv4i;   // LDS (__shared__)

// ---------------------------------------------------------------------------
// Projection: cam = (pos - t) @ R^T done as a 16x4 * 4x16 WMMA per tile.
// A (16x4, MxK)  = [x y z 1] per gaussian row.
// B (4x16, KxN)  : column j<3 holds {R[j][0], R[j][1], R[j][2], -(t . R_row_j)}.
// D (16x16) column j<3 = cam[:, j] for this 16-gaussian tile.
// One wave (32 threads) per tile; 32 tiles cover N=512.
// A/B VGPR layout per CDNA5 ISA 7.12.2: VGPR0 = K0 (lanes0-15) | K2 (lanes16-31),
// VGPR1 = K1 | K3. D: VGPR v -> M = v + (lane>=16 ? 8 : 0), N = lane & 15.
// ---------------------------------------------------------------------------
__global__ __launch_bounds__(32)
void sg_project_kernel(const float* __restrict__ pos,
                       const float* __restrict__ scl,
                       const float* __restrict__ col,
                       const float* __restrict__ opa,
                       const float* __restrict__ pose,
                       float* __restrict__ P,
                       float focal)
{
    __shared__ float cam[16 * 3];
    const int  lane = threadIdx.x;        // 0..31, full wave
    const int  m    = lane & 15;
    const int  g    = blockIdx.x * 16 + m;
    const bool hi   = lane >= 16;

    // pose is row-major 4x4: R[j][k] = pose[j*4+k], t[k] = pose[k*4+3]
    const float t0 = pose[3], t1 = pose[7], t2 = pose[11];

    // ---- A operand (branchless: EXEC must stay all-ones through the WMMA)
    const float px = pos[g*3+0], py = pos[g*3+1], pz = pos[g*3+2];
    v2f a;
    a.x = hi ? pz   : px;    // K=2 | K=0
    a.y = hi ? 1.0f : py;    // K=3 | K=1

    // ---- B operand: column n = lane&15, only n<3 meaningful
    const int   n  = m;
    const int   nn = (n < 3) ? n : 0;          // safe row index into pose
    const float bm = (n < 3) ? 1.0f : 0.0f;    // zero out unused columns
    const float r0 = pose[nn*4+0], r1 = pose[nn*4+1], r2 = pose[nn*4+2];
    const float w3 = -(r0*t0 + r1*t1 + r2*t2); // folds the (pos - t) translation
    v2f b;
    b.x = (hi ? r2 : r0) * bm;   // K=2 | K=0
    b.y = (hi ? w3 : r1) * bm;   // K=3 | K=1

#if __has_builtin(__builtin_amdgcn_wmma_f32_16x16x4_f32)
    v8f c = {0.f, 0.f, 0.f, 0.f, 0.f, 0.f, 0.f, 0.f};
    v8f d = __builtin_amdgcn_wmma_f32_16x16x4_f32(false, a, false, b,
                                                  (short)0, c, false, false);
    #pragma unroll
    for (int v = 0; v < 8; ++v) {
        if (n < 3) cam[(v + (hi ? 8 : 0)) * 3 + n] = d[v];
    }
#else
    if (!hi) {
        #pragma unroll
        for (int j = 0; j < 3; ++j)
            cam[m*3+j] = (px - t0) * pose[j*4+0]
                       + (py - t1) * pose[j*4+1]
                       + (pz - t2) * pose[j*4+2];
    }
#endif
    __syncthreads();

    if (!hi) {  // 16 lanes finalize 16 gaussians
        const float cx = cam[m*3+0], cy = cam[m*3+1], cz = cam[m*3+2];
        const float depth = fmaxf(-cz, 0.1f);
        const float invd  = focal / depth;
        const float sx    = cx * invd + (float)GW * 0.5f;
        const float sy    = cy * invd + (float)GH * 0.5f;
        const float smean = (scl[g*3+0] + scl[g*3+1] + scl[g*3+2]) * (1.0f/3.0f);
        const float ss    = fminf(fmaxf(smean * invd, 1.0f), 20.0f);
        const bool valid  = (sx >= 0.f) && (sx < (float)GW) &&
                            (sy >= 0.f) && (sy < (float)GH);
        P[0*GN+g] = sx;
        P[1*GN+g] = sy;
        P[2*GN+g] = floorf(sx);
        P[3*GN+g] = floorf(sy);
        P[4*GN+g] = floorf(ss * 3.0f);
        P[5*GN+g] = -0.5f / (ss * ss);
        P[6*GN+g] = valid ? opa[g] : 0.0f;   // invalid => alpha == 0 everywhere
        P[7*GN+g] = col[g*3+0];
        P[8*GN+g] = col[g*3+1];
        P[9*GN+g] = col[g*3+2];
    }
}

// ---------------------------------------------------------------------------
// Render: one thread per pixel; sequential fold over the 512 gaussians
// (compositing order matters). Parameter table (20 KB) staged into LDS via
// async global->LDS copies (ASYNCcnt path) when available.
// ---------------------------------------------------------------------------
__global__ __launch_bounds__(256)
void sg_render_kernel(const float* __restrict__ P, float* __restrict__ out)
{
    __shared__ float lds[NPAR * GN];            // 20480 bytes
    const int tid = threadIdx.x;
    const float4* gsrc = (const float4*)P;
    float4*       ldst = (float4*)lds;
    const int nvec = (NPAR * GN) / 4;           // 1280 float4s

#if __has_builtin(__builtin_amdgcn_global_load_async_to_lds_b128)
    for (int i = tid; i < nvec; i += 256) {
        __builtin_amdgcn_global_load_async_to_lds_b128(
            (gv4i*)(gsrc + i), (lv4i*)(ldst + i), 0, 0);
    }
    #if __has_builtin(__builtin_amdgcn_s_wait_asynccnt)
    __builtin_amdgcn_s_wait_asynccnt(0);        // wait this wave's async copies
    #else
    asm volatile("s_wait_asynccnt 0" ::: "memory");
    #endif
#else
    for (int i = tid; i < nvec; i += 256) ldst[i] = gsrc[i];
#endif
    __syncthreads();

    const int p = blockIdx.x * 256 + tid;
    if (p >= GW * GH) return;
    const float fx = (float)(p % GW);
    const float fy = (float)(p / GW);

    float r = 0.f, g = 0.f, bch = 0.f;
    for (int i = 0; i < GN; ++i) {
        const float xi  = lds[2*GN + i];
        const float yi  = lds[3*GN + i];
        const float rad = lds[4*GN + i];
        const float dx  = fx - lds[0*GN + i];
        const float dy  = fy - lds[1*GN + i];
        const float w   = __expf(lds[5*GN + i] * (dx*dx + dy*dy));
        const bool in_win = (fabsf(fx - xi) <= rad) && (fabsf(fy - yi) <= rad);
        const float alpha = in_win ? lds[6*GN + i] * w : 0.0f;
        const float om = 1.0f - alpha;
        r   = r   * om + lds[7*GN + i] * alpha;
        g   = g   * om + lds[8*GN + i] * alpha;
        bch = bch * om + lds[9*GN + i] * alpha;
    }
    out[0*GW*GH + p] = r;
    out[1*GW*GH + p] = g;
    out[2*GW*GH + p] = bch;
}

extern "C" void kernel_launch(void* const* d_in, const int* in_sizes, int n_in,
                              void* d_out, int out_size, void* d_ws, size_t ws_size,
                              hipStream_t stream)
{
    (void)in_sizes; (void)n_in; (void)out_size; (void)ws_size;
    const float* positions = (const float*)d_in[0];   // (512,3)
    const float* scales    = (const float*)d_in[1];   // (512,3)
    // d_in[2] rotations: unused by the reference
    const float* colors    = (const float*)d_in[3];   // (512,3)
    const float* opac      = (const float*)d_in[4];   // (512,)
    const float* pose      = (const float*)d_in[5];   // (4,4)

    float* P = (float*)d_ws;                          // 10*512 floats = 20 KB

    const double PI = 3.14159265358979323846;
    const float focal = (float)((double)GW / (2.0 * ::tan((55.0 * 0.5) * PI / 180.0)));

    sg_project_kernel<<<GN / 16, 32, 0, stream>>>(positions, scales, colors,
                                                  opac, pose, P, focal);
    sg_render_kernel<<<(GW * GH + 255) / 256, 256, 0, stream>>>(P, (float*)d_out);
}